// VariantEffectHead_87136296501260
// MI455X (gfx1250) — compile-verified
//
#include <hip/hip_runtime.h>
#include <math.h>

typedef __attribute__((ext_vector_type(2))) float v2f;
typedef __attribute__((ext_vector_type(8))) float v8f;

#define DIM 512
#define LSEQ 4096
#define WIN 32
#define LN_EPS 1e-5f

// ---------------------------------------------------------------------------
// Kernel 1: boundary-clamped window diff mean-pool + LayerNorm, all three
// scales. Grid: (B=16, S=3). 256 threads; each thread owns 2 channels.
// Writes the concatenated (16, 1536) activation matrix "cat" to workspace.
// ---------------------------------------------------------------------------
__global__ __launch_bounds__(256) void diff_pool_ln_kernel(
    const float* __restrict__ ref0, const float* __restrict__ ref1, const float* __restrict__ ref2,
    const float* __restrict__ alt0, const float* __restrict__ alt1, const float* __restrict__ alt2,
    const int*   __restrict__ vpos,
    const float* __restrict__ g0, const float* __restrict__ b0,
    const float* __restrict__ g1, const float* __restrict__ b1,
    const float* __restrict__ g2, const float* __restrict__ b2,
    float* __restrict__ cat)
{
    const int b = blockIdx.x;   // batch row
    const int s = blockIdx.y;   // scale (0=local, 1=regional, 2=macro)

    const float* ref   = (s == 0) ? ref0 : (s == 1) ? ref1 : ref2;
    const float* alt   = (s == 0) ? alt0 : (s == 1) ? alt1 : alt2;
    const float* gamma = (s == 0) ? g0   : (s == 1) ? g1   : g2;
    const float* beta  = (s == 0) ? b0   : (s == 1) ? b1   : b2;

    const int p     = vpos[b];
    const int start = (p - WIN < 0) ? 0 : (p - WIN);
    const int end   = (p + WIN > LSEQ - 1) ? (LSEQ - 1) : (p + WIN);
    const float inv_count = 1.0f / (float)(end - start + 1);

    const int c = threadIdx.x * 2;                 // channel pair owned by thread
    const float* rp = ref + ((size_t)b * LSEQ + start) * DIM + c;
    const float* ap = alt + ((size_t)b * LSEQ + start) * DIM + c;

    float acc0 = 0.0f, acc1 = 0.0f;
    for (int q = start; q <= end; ++q) {
        float2 rv = *(const float2*)rp;
        float2 av = *(const float2*)ap;
        acc0 += av.x - rv.x;
        acc1 += av.y - rv.y;
        rp += DIM; ap += DIM;
    }
    acc0 *= inv_count;
    acc1 *= inv_count;

    // LayerNorm over DIM=512 values (2 per thread) via LDS tree reduction.
    __shared__ float s_sum[256];
    __shared__ float s_sq[256];
    s_sum[threadIdx.x] = acc0 + acc1;
    s_sq[threadIdx.x]  = acc0 * acc0 + acc1 * acc1;
    __syncthreads();
    for (int off = 128; off > 0; off >>= 1) {
        if (threadIdx.x < (unsigned)off) {
            s_sum[threadIdx.x] += s_sum[threadIdx.x + off];
            s_sq[threadIdx.x]  += s_sq[threadIdx.x + off];
        }
        __syncthreads();
    }
    const float mu   = s_sum[0] * (1.0f / DIM);
    const float var  = s_sq[0]  * (1.0f / DIM) - mu * mu;
    const float rstd = rsqrtf(var + LN_EPS);

    float* out = cat + (size_t)b * (3 * DIM) + (size_t)s * DIM + c;
    out[0] = (acc0 - mu) * rstd * gamma[c]     + beta[c];
    out[1] = (acc1 - mu) * rstd * gamma[c + 1] + beta[c + 1];
}

// ---------------------------------------------------------------------------
// Kernel 2: M=16 GEMM on the WMMA pipe: out = act(A @ W + bias).
//   A: (16, K) row-major.  W: (K, N) row-major.  K, N, ACT compile-time.
// One wave per 16-wide N tile, K stepped by 4 via V_WMMA_F32_16X16X4_F32.
// Edge columns (N % 16 != 0) are handled by CLAMPING the column index: the
// clamped lanes compute garbage that lives only in their own output column,
// which is masked at store time — so the K-loop has zero EXEC manipulation.
// With N compile-time, W offsets are immediates (24-bit IOFFSET), so each
// unrolled step is: 1x global_load_b64 (A) + 2x global_load_b32 (W) + 1 wmma.
// ACT: 0 = none, 1 = exact GELU (erf), 2 = sigmoid.
// ---------------------------------------------------------------------------
template <int K, int N, int ACT>
__global__ __launch_bounds__(32) void gemm16_wmma_kernel(
    const float* __restrict__ A, const float* __restrict__ W,
    const float* __restrict__ bias,
    float* __restrict__ out, float* __restrict__ mirror)
{
    const int lane = threadIdx.x;
    const int half = lane >> 4;        // 0: K=k,k+1 | 1: K=k+2,k+3 (A frag)
    const int l15  = lane & 15;
    const int koff = half * 2;

    int n = blockIdx.x * 16 + l15;     // global output column
    bool nvalid = true;
    if constexpr ((N & 15) != 0) {
        nvalid = (n < N);
        if (!nvalid) n = N - 1;        // clamp: garbage stays in masked column
    }

    const float* __restrict__ aptr = A + (size_t)l15 * K + koff;
    const float* __restrict__ wptr = W + (size_t)koff * N + n;   // row k+koff, col n

    v8f acc = {};
    #pragma unroll 16
    for (int k = 0; k < K; k += 4) {
        v2f afrag = *(const v2f*)(aptr + k);   // A: K=k+koff, k+koff+1
        v2f bfrag;
        bfrag.x = wptr[(size_t)k * N];         // W[k+koff  ][n]  (immediate offset)
        bfrag.y = wptr[(size_t)k * N + N];     // W[k+koff+1][n]  (immediate offset)
        acc = __builtin_amdgcn_wmma_f32_16x16x4_f32(
            /*neg_a=*/false, afrag, /*neg_b=*/false, bfrag,
            /*c_mod=*/(short)0, acc, /*reuse_a=*/false, /*reuse_b=*/false);
    }

    if (nvalid) {
        const float bv = bias[n];
        #pragma unroll
        for (int r = 0; r < 8; ++r) {
            const int row = r + half * 8;      // C/D layout: lanes>=16 hold M=r+8
            float v = acc[r] + bv;
            if constexpr (ACT == 1) {
                v = 0.5f * v * (1.0f + erff(v * 0.70710678118654752f));
            } else if constexpr (ACT == 2) {
                v = 1.0f / (1.0f + __expf(-v));
            }
            out[(size_t)row * N + n] = v;
            if (mirror) mirror[(size_t)row * N + n] = v;
        }
    }
}

// ---------------------------------------------------------------------------
// Host-side launch. Input order follows setup_inputs():
//  0-5: ref/alt x local/regional/macro   6: variant_pos (int32)
//  7-12: ln gamma/beta x3
//  13: W_fuse 14: b_fuse 15: W_i1 16: b_i1 17: W_i2 18: b_i2
//  19: W_t1 20: b_t1 21: W_t2 22: b_t2 23: W_m1 24: b_m1 25: W_m2 26: b_m2
// Output (flat): impact(16) | tissue(16x200) | mech(16x6) | fused(16x512)
// ---------------------------------------------------------------------------
extern "C" void kernel_launch(void* const* d_in, const int* in_sizes, int n_in,
                              void* d_out, int out_size, void* d_ws, size_t ws_size,
                              hipStream_t stream) {
    const float* ref_l = (const float*)d_in[0];
    const float* ref_r = (const float*)d_in[1];
    const float* ref_m = (const float*)d_in[2];
    const float* alt_l = (const float*)d_in[3];
    const float* alt_r = (const float*)d_in[4];
    const float* alt_m = (const float*)d_in[5];
    const int*   vpos  = (const int*)d_in[6];
    const float* g_l = (const float*)d_in[7];
    const float* b_l = (const float*)d_in[8];
    const float* g_r = (const float*)d_in[9];
    const float* b_r = (const float*)d_in[10];
    const float* g_m = (const float*)d_in[11];
    const float* b_m = (const float*)d_in[12];
    const float* W_fuse = (const float*)d_in[13]; const float* bb_fuse = (const float*)d_in[14];
    const float* W_i1   = (const float*)d_in[15]; const float* bb_i1   = (const float*)d_in[16];
    const float* W_i2   = (const float*)d_in[17]; const float* bb_i2   = (const float*)d_in[18];
    const float* W_t1   = (const float*)d_in[19]; const float* bb_t1   = (const float*)d_in[20];
    const float* W_t2   = (const float*)d_in[21]; const float* bb_t2   = (const float*)d_in[22];
    const float* W_m1   = (const float*)d_in[23]; const float* bb_m1   = (const float*)d_in[24];
    const float* W_m2   = (const float*)d_in[25]; const float* bb_m2   = (const float*)d_in[26];

    // Workspace layout (floats)
    float* ws    = (float*)d_ws;
    float* cat   = ws;              // 16 x 1536 = 24576
    float* fused = ws + 24576;      // 16 x 512  =  8192
    float* h_t   = ws + 32768;      // 16 x 512  =  8192
    float* h_i   = ws + 40960;      // 16 x 256  =  4096
    float* h_m   = ws + 45056;      // 16 x 256  =  4096

    // Output layout (floats)
    float* o        = (float*)d_out;
    float* o_impact = o;            // (16,)
    float* o_tissue = o + 16;       // (16, 200)
    float* o_mech   = o + 3216;     // (16, 6)
    float* o_fused  = o + 3312;     // (16, 512)

    // 1) diff-pool + LayerNorm -> cat (16, 1536)
    dim3 gp(16, 3);
    diff_pool_ln_kernel<<<gp, 256, 0, stream>>>(
        ref_l, ref_r, ref_m, alt_l, alt_r, alt_m, vpos,
        g_l, b_l, g_r, b_r, g_m, b_m, cat);

    // 2) fused = gelu(cat @ W_fuse + b_fuse)  (16,1536)x(1536,512)
    gemm16_wmma_kernel<1536, 512, 1><<<32, 32, 0, stream>>>(cat, W_fuse, bb_fuse, fused, o_fused);

    // 3) impact head: h_i = gelu(fused@W_i1+b), impact = sigmoid(h_i@W_i2+b)
    gemm16_wmma_kernel<512, 256, 1><<<16, 32, 0, stream>>>(fused, W_i1, bb_i1, h_i, nullptr);
    gemm16_wmma_kernel<256, 1, 2><<<1, 32, 0, stream>>>(h_i, W_i2, bb_i2, o_impact, nullptr);

    // 4) tissue head: h_t = gelu(fused@W_t1+b), tissue = h_t@W_t2+b
    gemm16_wmma_kernel<512, 512, 1><<<32, 32, 0, stream>>>(fused, W_t1, bb_t1, h_t, nullptr);
    gemm16_wmma_kernel<512, 200, 0><<<13, 32, 0, stream>>>(h_t, W_t2, bb_t2, o_tissue, nullptr);

    // 5) mechanism head: h_m = gelu(fused@W_m1+b), mech = h_m@W_m2+b
    gemm16_wmma_kernel<512, 256, 1><<<16, 32, 0, stream>>>(fused, W_m1, bb_m1, h_m, nullptr);
    gemm16_wmma_kernel<256, 6, 0><<<1, 32, 0, stream>>>(h_m, W_m2, bb_m2, o_mech, nullptr);
}